// FNO2d_89154931130979
// MI455X (gfx1250) — compile-verified
//
#include <hip/hip_runtime.h>
#include <math.h>

// FNO2d for MI455X / gfx1250 (wave32, WMMA f32 16x16x4).
// Spectral conv implemented as truncated DFT GEMMs (24x12 kept modes),
// inverse-y DFT fused with 1x1 conv + bias + exact GELU in one WMMA kernel.
// All WMMA K-loops are branch-free: DFT tables are zero-padded so edge
// guards vanish (out-of-range coefficient rows are 0).

#define OD   265     // padded spatial size (256 + 9)
#define SS   256
#define NBATCH 8
#define CC   32      // WIDTH
#define MD   12      // MODES
#define K24  24      // 2*MODES
#define FYS  268     // FY table stride (OD rounded to 4)
#define EYR  272     // EY padded rows (17*16)
#define BNPARTS 512

typedef __attribute__((ext_vector_type(2))) float v2f;
typedef __attribute__((ext_vector_type(8))) float v8f;

__device__ __forceinline__ float gelu_exact(float v) {
    return 0.5f * v * (1.0f + erff(v * 0.70710678118654752440f));
}

__device__ __forceinline__ v8f wmma_f32(v2f a, v2f b, v8f c) {
    return __builtin_amdgcn_wmma_f32_16x16x4_f32(false, a, false, b, (short)0, c, false, false);
}

__device__ __forceinline__ v2f ldv2(const float* p) {
    return *(const v2f*)p;
}

// ---------------- DFT coefficient tables (built per launch) ----------------
// FY32 [32][268]: forward-y rows, zero-padded. k<12: cos(2pi k y/N);
//                 12<=k<24: -sin(2pi (k-12) y/N); rows>=24 or y>=265: 0.
// CXF/SXF [24][265]: forward-x cos/sin at freq f_j = j (j<12) or 253+(j-12)
// CIX/SIX [24][265]: inverse-x cos/sin at freq j (irfft zero-pads at end)
// EY [272][24]: inverse-y (irfft along last axis) with 1/N^2 folded in;
//               rows y>=265 are zero (edge-tile padding).
__global__ void init_tables_kernel(float* __restrict__ FY32, float* __restrict__ CXF,
                                   float* __restrict__ SXF, float* __restrict__ CIX,
                                   float* __restrict__ SIX, float* __restrict__ EY) {
    int t = blockIdx.x * blockDim.x + threadIdx.x;
    const double TWO_PI = 6.283185307179586476925286766559;
    const double invN = 1.0 / (double)OD;
    if (t < 32 * FYS) {
        int k = t / FYS;
        int p = t % FYS;
        float v = 0.f;
        if (k < K24 && p < OD) {
            if (k < MD) {
                long long m = ((long long)k * p) % OD;
                v = (float)cos(TWO_PI * (double)m * invN);
            } else {
                long long m = ((long long)(k - MD) * p) % OD;
                v = (float)(-sin(TWO_PI * (double)m * invN));
            }
        }
        FY32[t] = v;
    }
    if (t < EYR * K24) {
        int y = t / K24;
        int k = t % K24;
        double e = 0.0;
        if (y < OD) {
            double inv2 = invN * invN;
            if (k == 0) {
                e = inv2;
            } else if (k < MD) {
                long long mm = ((long long)k * y) % OD;
                e = 2.0 * inv2 * cos(TWO_PI * (double)mm * invN);
            } else if (k < 2 * MD - 1) {
                long long mm = ((long long)(k - (MD - 1)) * y) % OD;
                e = -2.0 * inv2 * sin(TWO_PI * (double)mm * invN);
            }
        }
        EY[t] = (float)e;
    }
    if (t < K24 * OD) {
        int k = t / OD;
        int p = t % OD;
        int f = (k < MD) ? k : (OD - MD + (k - MD));     // 253..264 for bottom block
        long long mf = ((long long)f * p) % OD;
        double af = TWO_PI * (double)mf * invN;
        CXF[t] = (float)cos(af);
        SXF[t] = (float)sin(af);
        long long mi = ((long long)k * p) % OD;
        double ai = TWO_PI * (double)mi * invN;
        CIX[t] = (float)cos(ai);
        SIX[t] = (float)sin(ai);
    }
}

// ---------------- fc0: lift 3 channels + grid to 32, zero-pad to 265x265 ----------
// h layout: [b][x][y][c] with c contiguous.
__global__ void fc0_kernel(const float* __restrict__ xin, const float* __restrict__ w,
                           const float* __restrict__ bias, float* __restrict__ h) {
    int t = blockIdx.x * blockDim.x + threadIdx.x;
    if (t >= NBATCH * OD * OD * CC) return;
    int c = t & 31;
    int q = t >> 5;
    int y = q % OD; q /= OD;
    int x = q % OD;
    int b = q / OD;
    float v = 0.f;
    if (x < SS && y < SS) {
        int base = ((b * 3) * SS + x) * SS + y;
        v = bias[c];
        v += xin[base] * w[c];
        v += xin[base + SS * SS] * w[32 + c];
        v += xin[base + 2 * SS * SS] * w[64 + c];
        v += (x * (1.0f / 255.0f)) * w[96 + c];
        v += (y * (1.0f / 255.0f)) * w[128 + c];
    }
    h[t] = v;
}

// ---------------- forward y-DFT: T[b][x][k24][c] = sum_y FY[k][y] * h[b][x][y][c] ----
// One wave per (b,x): GEMM M=24(pad 32), K=265(pad 268), N=32. Branch-free K loop:
// FY32 is zero for y>=265 and rows>=24, so B loads may read (in-bounds) garbage.
__global__ __launch_bounds__(32)
void ydft_wmma_kernel(const float* __restrict__ h, const float* __restrict__ FY32,
                      float* __restrict__ T) {
    int x = blockIdx.x, b = blockIdx.y;
    int lane = threadIdx.x & 31;
    int half = lane >> 4, l16 = lane & 15;
    const float* Hx = h + ((b * OD + x) * OD) * CC;
    const float* fyr0 = FY32 + l16 * FYS;
    const float* fyr1 = FY32 + (16 + l16) * FYS;
    v8f acc00 = {}, acc01 = {}, acc10 = {}, acc11 = {};
    for (int kk = 0; kk < FYS; kk += 4) {
        int k0 = kk + 2 * half;                  // even
        v2f a0 = ldv2(fyr0 + k0);
        v2f a1 = ldv2(fyr1 + k0);
        v2f b0, b1;
        #pragma unroll
        for (int j = 0; j < 2; ++j) {
            b0[j] = Hx[(k0 + j) * CC + l16];
            b1[j] = Hx[(k0 + j) * CC + 16 + l16];
        }
        acc00 = wmma_f32(a0, b0, acc00);
        acc01 = wmma_f32(a0, b1, acc01);
        acc10 = wmma_f32(a1, b0, acc10);
        acc11 = wmma_f32(a1, b1, acc11);
    }
    float* Tx = T + ((b * OD + x) * K24) * CC;
    #pragma unroll
    for (int j = 0; j < 8; ++j) {
        int m = j + 8 * half;
        Tx[m * CC + l16] = acc00[j];
        Tx[m * CC + 16 + l16] = acc01[j];
        int m1 = 16 + m;
        if (m1 < K24) {
            Tx[m1 * CC + l16] = acc10[j];
            Tx[m1 * CC + 16 + l16] = acc11[j];
        }
    }
}

// ---------------- forward x-DFT (small): X[b][j][ky][re/im][c] ----------------
__global__ void xdft_kernel(const float* __restrict__ T, const float* __restrict__ CXF,
                            const float* __restrict__ SXF, float* __restrict__ X) {
    int t = blockIdx.x * blockDim.x + threadIdx.x;
    if (t >= NBATCH * K24 * MD * CC) return;
    int c = t & 31;
    int q = t >> 5;
    int ky = q % MD; q /= MD;
    int j = q % K24;
    int b = q / K24;
    const float* Cj = CXF + j * OD;
    const float* Sj = SXF + j * OD;
    float re = 0.f, im = 0.f;
    for (int x = 0; x < OD; ++x) {
        const float* Tp = T + ((b * OD + x) * K24) * CC;
        float tr = Tp[ky * CC + c];
        float ti = Tp[(MD + ky) * CC + c];
        float cf = Cj[x], sf = Sj[x];
        re += cf * tr + sf * ti;        // e^{-i th}(tr + i ti)
        im += cf * ti - sf * tr;
    }
    float* Xp = X + (((b * K24 + j) * MD + ky) * 2) * CC;
    Xp[c] = re;
    Xp[CC + c] = im;
}

// ---------------- complex mode mixing: Y = X * Wc (32x32 per mode) --------------
__global__ void mix_kernel(const float* __restrict__ X, const float* __restrict__ w1,
                           const float* __restrict__ w2, float* __restrict__ Y) {
    int t = blockIdx.x * blockDim.x + threadIdx.x;
    if (t >= NBATCH * K24 * MD * CC) return;
    int o = t & 31;
    int q = t >> 5;
    int ky = q % MD; q /= MD;
    int j = q % K24;
    int b = q / K24;
    const float* W = (j < MD) ? w1 : w2;
    int jx = (j < MD) ? j : j - MD;
    const float* Xp = X + (((b * K24 + j) * MD + ky) * 2) * CC;
    float yre = 0.f, yim = 0.f;
    for (int i = 0; i < CC; ++i) {
        float xr = Xp[i], xi = Xp[CC + i];
        int wbase = (((i * CC + o) * MD + jx) * MD + ky) * 2;
        float wr = W[wbase], wi = W[wbase + 1];
        yre += xr * wr - xi * wi;
        yim += xr * wi + xi * wr;
    }
    float* Yp = Y + (((b * K24 + j) * MD + ky) * 2) * CC;
    Yp[o] = yre;
    Yp[CC + o] = yim;
}

// ---------------- inverse x-DFT (small), writing directly the WMMA B-matrix ----
// B1[b][x][24][32]: rows 0..11 = Re(ky), rows 12..22 = Im(ky=1..11), row 23 = 0.
__global__ void invx_kernel(const float* __restrict__ Y, const float* __restrict__ CIX,
                            const float* __restrict__ SIX, float* __restrict__ B1) {
    int t = blockIdx.x * blockDim.x + threadIdx.x;
    if (t >= NBATCH * OD * MD * CC) return;
    int c = t & 31;
    int q = t >> 5;
    int ky = q % MD; q /= MD;
    int x = q % OD;
    int b = q / OD;
    float gre = 0.f, gim = 0.f;
    for (int j = 0; j < K24; ++j) {
        const float* Yp = Y + (((b * K24 + j) * MD + ky) * 2) * CC;
        float yr = Yp[c], yi = Yp[CC + c];
        float cf = CIX[j * OD + x], sf = SIX[j * OD + x];
        gre += cf * yr - sf * yi;       // e^{+i th}(yr + i yi)
        gim += sf * yr + cf * yi;
    }
    float* Bp = B1 + ((b * OD + x) * K24) * CC;
    Bp[ky * CC + c] = gre;
    if (ky >= 1) Bp[(11 + ky) * CC + c] = gim;   // rows 12..22
    else         Bp[23 * CC + c] = 0.f;          // zero pad row
}

// -------- fused inverse-y DFT + 1x1 conv + bias + (gelu): two WMMA GEMMs ------
// One wave per (b, x, 16-row tile of y). GEMM1: EY(272x24)*B1x(24x32),
// GEMM2: Hx(~272x32)*W(32x32). Branch-free K loops (EY rows >=265 are zero;
// Hx over-reads land in the adjacent workspace buffer and are never stored).
__global__ __launch_bounds__(32)
void invy_conv_wmma_kernel(const float* __restrict__ B1, const float* __restrict__ EY,
                           const float* __restrict__ h, const float* __restrict__ W,
                           const float* __restrict__ bias, float* __restrict__ hout,
                           int apply_gelu) {
    int mt = blockIdx.x, x = blockIdx.y, b = blockIdx.z;
    int lane = threadIdx.x & 31;
    int half = lane >> 4, l16 = lane & 15;
    int m0 = mt * 16;
    int yA = m0 + l16;                            // up to 271, EY padded
    v8f acc0 = {}, acc1 = {};
    const float* B1x = B1 + ((b * OD + x) * K24) * CC;
    const float* eyr = EY + yA * K24;
    // GEMM1: spectral inverse along y, K = 24
    for (int kk = 0; kk < K24; kk += 4) {
        int k0 = kk + 2 * half;
        v2f a = ldv2(eyr + k0);
        v2f b0, b1;
        #pragma unroll
        for (int j = 0; j < 2; ++j) {
            b0[j] = B1x[(k0 + j) * CC + l16];
            b1[j] = B1x[(k0 + j) * CC + 16 + l16];
        }
        acc0 = wmma_f32(a, b0, acc0);
        acc1 = wmma_f32(a, b1, acc1);
    }
    // GEMM2: 1x1 conv, K = 32
    const float* Hrow = h + (((b * OD + x) * OD) + yA) * CC;
    for (int kk = 0; kk < CC; kk += 4) {
        int k0 = kk + 2 * half;
        v2f a = ldv2(Hrow + k0);
        v2f b0, b1;
        #pragma unroll
        for (int j = 0; j < 2; ++j) {
            b0[j] = W[(k0 + j) * CC + l16];
            b1[j] = W[(k0 + j) * CC + 16 + l16];
        }
        acc0 = wmma_f32(a, b0, acc0);
        acc1 = wmma_f32(a, b1, acc1);
    }
    float bia0 = bias[l16], bia1 = bias[16 + l16];
    float* Ox = hout + ((b * OD + x) * OD) * CC;
    #pragma unroll
    for (int j = 0; j < 8; ++j) {
        int yy = m0 + j + 8 * half;
        if (yy < OD) {
            float v0 = acc0[j] + bia0;
            float v1 = acc1[j] + bia1;
            if (apply_gelu) { v0 = gelu_exact(v0); v1 = gelu_exact(v1); }
            Ox[yy * CC + l16] = v0;
            Ox[yy * CC + 16 + l16] = v1;
        }
    }
}

// -------- fused fc1 (WMMA, 32->128) + exact gelu + fc2 (128->3) --------------
// One wave per 16 pixels (contiguous y within a row; crop region only).
__global__ __launch_bounds__(32)
void fc12_wmma_kernel(const float* __restrict__ h, const float* __restrict__ w1,
                      const float* __restrict__ b1, const float* __restrict__ w2,
                      const float* __restrict__ b2, float* __restrict__ out) {
    __shared__ float tbuf[16][129];
    int lane = threadIdx.x & 31;
    int half = lane >> 4, l16 = lane & 15;
    int p0 = blockIdx.x * 16;
    int b = p0 >> 16;
    int rem = p0 & 65535;
    int xx = rem >> 8;
    int yy0 = rem & 255;
    const float* Hrow = h + ((b * OD + xx) * OD + yy0 + l16) * CC;
    v8f acc[8];
    #pragma unroll
    for (int nt = 0; nt < 8; ++nt) acc[nt] = (v8f){};
    for (int kk = 0; kk < CC; kk += 4) {
        int k0 = kk + 2 * half;
        v2f a = ldv2(Hrow + k0);
        #pragma unroll
        for (int nt = 0; nt < 8; ++nt) {
            v2f bb;
            #pragma unroll
            for (int j = 0; j < 2; ++j) {
                bb[j] = w1[(k0 + j) * 128 + nt * 16 + l16];
            }
            acc[nt] = wmma_f32(a, bb, acc[nt]);
        }
    }
    #pragma unroll
    for (int nt = 0; nt < 8; ++nt) {
        int n = nt * 16 + l16;
        float bv = b1[n];
        #pragma unroll
        for (int j = 0; j < 8; ++j) {
            int m = j + 8 * half;
            tbuf[m][n] = gelu_exact(acc[nt][j] + bv);
        }
    }
    __syncthreads();
    for (int idx = lane; idx < 48; idx += 32) {
        int p = idx / 3, o = idx % 3;
        float s = b2[o];
        for (int hh = 0; hh < 128; ++hh) s += tbuf[p][hh] * w2[hh * 3 + o];
        out[((b * 3 + o) * SS + xx) * SS + yy0 + p] = s;
    }
}

// ---------------- batchnorm: deterministic two-stage reduction ----------------
__global__ void bn_partial_kernel(const float* __restrict__ h2, float* __restrict__ red) {
    int o = blockIdx.y;
    int part = blockIdx.x;
    const int total = NBATCH * SS * SS;              // 524288 per channel
    int chunk = (total + BNPARTS - 1) / BNPARTS;     // 1024
    int start = part * chunk;
    int end = start + chunk; if (end > total) end = total;
    float s = 0.f, s2 = 0.f;
    for (int i = start + (int)threadIdx.x; i < end; i += 256) {
        int b = i >> 16;
        int rem = i & 65535;
        float v = h2[((b * 3 + o) << 16) + rem];
        s += v; s2 += v * v;
    }
    __shared__ float sh[256], sh2[256];
    sh[threadIdx.x] = s; sh2[threadIdx.x] = s2;
    __syncthreads();
    for (int st = 128; st > 0; st >>= 1) {
        if ((int)threadIdx.x < st) {
            sh[threadIdx.x] += sh[threadIdx.x + st];
            sh2[threadIdx.x] += sh2[threadIdx.x + st];
        }
        __syncthreads();
    }
    if (threadIdx.x == 0) {
        red[(o * BNPARTS + part) * 2] = sh[0];
        red[(o * BNPARTS + part) * 2 + 1] = sh2[0];
    }
}

__global__ void bn_final_kernel(const float* __restrict__ red, const float* __restrict__ g,
                                const float* __restrict__ bb, float* __restrict__ stats) {
    int o = threadIdx.x;
    if (o < 3) {
        float s = 0.f, s2 = 0.f;
        for (int p = 0; p < BNPARTS; ++p) {
            s += red[(o * BNPARTS + p) * 2];
            s2 += red[(o * BNPARTS + p) * 2 + 1];
        }
        float n = (float)(NBATCH * SS * SS);
        float mean = s / n;
        float var = s2 / n - mean * mean;
        float sc = g[o] * rsqrtf(var + 1e-5f);
        stats[o * 2] = sc;
        stats[o * 2 + 1] = bb[o] - mean * sc;
    }
}

__global__ void bn_apply_kernel(const float* __restrict__ h2, const float* __restrict__ stats,
                                float* __restrict__ out) {
    int idx = blockIdx.x * blockDim.x + threadIdx.x;
    if (idx >= NBATCH * 3 * SS * SS) return;
    int o = (idx >> 16) % 3;
    out[idx] = h2[idx] * stats[o * 2] + stats[o * 2 + 1];
}

// =============================== launch ======================================
extern "C" void kernel_launch(void* const* d_in, const int* in_sizes, int n_in,
                              void* d_out, int out_size, void* d_ws, size_t ws_size,
                              hipStream_t stream) {
    const float* xin   = (const float*)d_in[0];
    const float* fc0_w = (const float*)d_in[1];
    const float* fc0_b = (const float*)d_in[2];
    const float* scw1[4], *scw2[4], *ww[4], *wb[4];
    for (int l = 0; l < 4; ++l) {       // dict order: sc_w1, sc_w2, w_w, w_b per layer
        scw1[l] = (const float*)d_in[3 + 4 * l];
        scw2[l] = (const float*)d_in[4 + 4 * l];
        ww[l]   = (const float*)d_in[5 + 4 * l];
        wb[l]   = (const float*)d_in[6 + 4 * l];
    }
    const float* fc1_w = (const float*)d_in[19];
    const float* fc1_b = (const float*)d_in[20];
    const float* fc2_w = (const float*)d_in[21];
    const float* fc2_b = (const float*)d_in[22];
    const float* bn_g  = (const float*)d_in[23];
    const float* bn_b  = (const float*)d_in[24];

    float* ws = (float*)d_ws;
    const size_t NH  = (size_t)NBATCH * OD * OD * CC;       // 17,977,600
    const size_t NT  = (size_t)NBATCH * OD * K24 * CC;      // 1,628,160
    const size_t NX  = (size_t)NBATCH * K24 * MD * 2 * CC;  // 147,456
    const size_t NB1 = (size_t)NBATCH * OD * K24 * CC;      // 1,628,160
    // Order matters: hA, hB, Tb adjacent so small WMMA over-reads past a
    // ping-pong buffer stay inside the workspace.
    float* hA  = ws;
    float* hB  = hA + NH;
    float* Tb  = hB + NH;
    float* Xb  = Tb + NT;
    float* Yb  = Xb + NX;
    float* B1b = Yb + NX;
    float* FY32 = B1b + NB1;
    float* CXF = FY32 + (size_t)32 * FYS;
    float* SXF = CXF + (size_t)K24 * OD;
    float* CIX = SXF + (size_t)K24 * OD;
    float* SIX = CIX + (size_t)K24 * OD;
    float* EY  = SIX + (size_t)K24 * OD;
    float* h2  = EY + (size_t)EYR * K24;
    float* red = h2 + (size_t)NBATCH * 3 * SS * SS;
    float* stats = red + (size_t)3 * BNPARTS * 2;

    init_tables_kernel<<<(32 * FYS + 255) / 256, 256, 0, stream>>>(FY32, CXF, SXF, CIX, SIX, EY);

    fc0_kernel<<<(NBATCH * OD * OD * CC + 255) / 256, 256, 0, stream>>>(xin, fc0_w, fc0_b, hA);

    float* hin = hA;
    float* hout = hB;
    for (int l = 0; l < 4; ++l) {
        ydft_wmma_kernel<<<dim3(OD, NBATCH), 32, 0, stream>>>(hin, FY32, Tb);
        xdft_kernel<<<(NBATCH * K24 * MD * CC + 255) / 256, 256, 0, stream>>>(Tb, CXF, SXF, Xb);
        mix_kernel<<<(NBATCH * K24 * MD * CC + 255) / 256, 256, 0, stream>>>(Xb, scw1[l], scw2[l], Yb);
        invx_kernel<<<(NBATCH * OD * MD * CC + 255) / 256, 256, 0, stream>>>(Yb, CIX, SIX, B1b);
        invy_conv_wmma_kernel<<<dim3(17, OD, NBATCH), 32, 0, stream>>>(
            B1b, EY, hin, ww[l], wb[l], hout, (l < 3) ? 1 : 0);
        float* tmp = hin; hin = hout; hout = tmp;
    }

    fc12_wmma_kernel<<<(NBATCH * SS * SS) / 16, 32, 0, stream>>>(hin, fc1_w, fc1_b, fc2_w, fc2_b, h2);

    bn_partial_kernel<<<dim3(BNPARTS, 3), 256, 0, stream>>>(h2, red);
    bn_final_kernel<<<1, 64, 0, stream>>>(red, bn_g, bn_b, stats);
    bn_apply_kernel<<<(NBATCH * 3 * SS * SS + 255) / 256, 256, 0, stream>>>(h2, stats, (float*)d_out);
}